// DeformationTrajectoryAttentionBlock_40664750359304
// MI455X (gfx1250) — compile-verified
//
#include <hip/hip_runtime.h>
#include <math.h>

#define DMODEL 384
#define NTOK   1024
#define SWIN   256
#define FNUM   4
#define HNUM   6
#define HDIM   64
#define SCALEF 0.125f
#define EPSV   1e-5f
#define QKVLD  1152

typedef __attribute__((ext_vector_type(2))) float v2f;
typedef __attribute__((ext_vector_type(8))) float v8f;

// ---------------- patch embed: 2x2 maxpool, (BF,D,32,32) -> tok (N,D) ----------------
__global__ void patch_kernel(const float* __restrict__ x, float* __restrict__ tok) {
    int i = blockIdx.x * 256 + threadIdx.x;
    if (i >= NTOK * DMODEL) return;
    int n = i / DMODEL, d = i % DMODEL;
    int f = n >> 8, r = n & 255, gy = r >> 4, gx = r & 15;
    const float* p = x + ((((long)f * DMODEL + d) * 32 + gy * 2) * 32 + gx * 2);
    tok[i] = fmaxf(fmaxf(p[0], p[1]), fmaxf(p[32], p[33]));
}

// ---------------- layernorm over D per row ----------------
__global__ __launch_bounds__(DMODEL) void ln_kernel(const float* __restrict__ in, float* __restrict__ out,
                                                    const float* __restrict__ g, const float* __restrict__ b) {
    __shared__ float red[DMODEL];
    int n = blockIdx.x, t = threadIdx.x;
    float v = in[(long)n * DMODEL + t];
    red[t] = v; __syncthreads();
    for (int st = 256; st > 0; st >>= 1) { if (t < st && t + st < DMODEL) red[t] += red[t + st]; __syncthreads(); }
    float mean = red[0] * (1.0f / DMODEL); __syncthreads();
    float dv = v - mean;
    red[t] = dv * dv; __syncthreads();
    for (int st = 256; st > 0; st >>= 1) { if (t < st && t + st < DMODEL) red[t] += red[t + st]; __syncthreads(); }
    float var = red[0] * (1.0f / DMODEL);
    out[(long)n * DMODEL + t] = dv * rsqrtf(var + EPSV) * g[t] + b[t];
}

// ---------------- generic f32 WMMA GEMM: C = act(alpha*A@B + bias + resid) ----------------
// one wave -> one 32x32 C tile: 2 A-frags x 2 B-frags -> 4 v_wmma_f32_16x16x4_f32 per k-step
template<int TRANSB>
__global__ __launch_bounds__(256) void gemm_wmma_f32(
    const float* __restrict__ A, int lda, long strideA,
    const float* __restrict__ B, int ldb, long strideB,
    float* __restrict__ C, int ldc, long strideC,
    const float* __restrict__ bias, const float* __restrict__ resid,
    int M, int N, int K, float alpha, int act)
{
    int z = blockIdx.z;
    A += (long)z * strideA; B += (long)z * strideB; C += (long)z * strideC;
    int wid  = threadIdx.x >> 5;
    int lane = threadIdx.x & 31;
    int ntile = N >> 5;
    int tiles = (M >> 5) * ntile;
    int tile = blockIdx.x * 8 + wid;
    if (tile >= tiles) return;                 // wave-uniform: EXEC stays all-ones for WMMA
    int m0 = (tile / ntile) << 5;
    int n0 = (tile % ntile) << 5;
    int l16 = lane & 15;
    int koff = (lane < 16) ? 0 : 2;

    // strength-reduced pointers
    const float* ap0 = A + (long)(m0 + l16) * lda + koff;
    const float* ap1 = ap0 + (long)16 * lda;
    const float* bp0;
    const float* bp1;
    if (TRANSB) {                                // B element (k,c) = B[c*ldb + k]
        bp0 = B + (long)(n0 + l16) * ldb + koff;
        bp1 = bp0 + (long)16 * ldb;
    } else {                                     // B element (k,c) = B[k*ldb + c]
        bp0 = B + (long)koff * ldb + n0 + l16;
        bp1 = bp0 + 16;
    }

    v8f acc00 = {0.f,0.f,0.f,0.f,0.f,0.f,0.f,0.f};
    v8f acc01 = acc00, acc10 = acc00, acc11 = acc00;

    #pragma unroll 2
    for (int k0 = 0; k0 < K; k0 += 4) {
        v2f a0, a1, b0, b1;
        a0.x = ap0[0]; a0.y = ap0[1];
        a1.x = ap1[0]; a1.y = ap1[1];
        if (TRANSB) {
            b0.x = bp0[0]; b0.y = bp0[1];
            b1.x = bp1[0]; b1.y = bp1[1];
            bp0 += 4; bp1 += 4;
        } else {
            b0.x = bp0[0]; b0.y = bp0[ldb];
            b1.x = bp1[0]; b1.y = bp1[ldb];
            bp0 += (long)4 * ldb; bp1 += (long)4 * ldb;
        }
        ap0 += 4; ap1 += 4;
        acc00 = __builtin_amdgcn_wmma_f32_16x16x4_f32(false, a0, false, b0, (short)0, acc00, false, false);
        acc01 = __builtin_amdgcn_wmma_f32_16x16x4_f32(false, a0, false, b1, (short)0, acc01, false, false);
        acc10 = __builtin_amdgcn_wmma_f32_16x16x4_f32(false, a1, false, b0, (short)0, acc10, false, false);
        acc11 = __builtin_amdgcn_wmma_f32_16x16x4_f32(false, a1, false, b1, (short)0, acc11, false, false);
    }

    int colA = n0 + l16, colB = colA + 16;
    int rb0 = m0 + ((lane < 16) ? 0 : 8);
    int rb1 = rb0 + 16;
    float biasA = bias ? bias[colA] : 0.f;
    float biasB = bias ? bias[colB] : 0.f;
    #pragma unroll
    for (int r = 0; r < 8; r++) {
        float v00 = acc00[r] * alpha + biasA;
        float v01 = acc01[r] * alpha + biasB;
        float v10 = acc10[r] * alpha + biasA;
        float v11 = acc11[r] * alpha + biasB;
        if (resid) {
            v00 += resid[(long)(rb0 + r) * ldc + colA];
            v01 += resid[(long)(rb0 + r) * ldc + colB];
            v10 += resid[(long)(rb1 + r) * ldc + colA];
            v11 += resid[(long)(rb1 + r) * ldc + colB];
        }
        if (act) {
            v00 = 0.5f * v00 * (1.0f + erff(v00 * 0.70710678118654752f));
            v01 = 0.5f * v01 * (1.0f + erff(v01 * 0.70710678118654752f));
            v10 = 0.5f * v10 * (1.0f + erff(v10 * 0.70710678118654752f));
            v11 = 0.5f * v11 * (1.0f + erff(v11 * 0.70710678118654752f));
        }
        C[(long)(rb0 + r) * ldc + colA] = v00;
        C[(long)(rb0 + r) * ldc + colB] = v01;
        C[(long)(rb1 + r) * ldc + colA] = v10;
        C[(long)(rb1 + r) * ldc + colB] = v11;
    }
}

// ---------------- frame softmax: attnw[h,n,s,f] = softmax_f(SCALE*qk[h,n,f*S+s]) ----------------
__global__ void tsoftmax_kernel(const float* __restrict__ qk, float* __restrict__ aw) {
    int i = blockIdx.x * 256 + threadIdx.x;
    if (i >= HNUM * NTOK * SWIN) return;
    int h = i >> 18; int r = i & 262143; int n = r >> 8; int s = r & 255;
    const float* q = qk + ((long)h * NTOK + n) * NTOK + s;
    float v0 = SCALEF * q[0], v1 = SCALEF * q[256], v2 = SCALEF * q[512], v3 = SCALEF * q[768];
    float m = fmaxf(fmaxf(v0, v1), fmaxf(v2, v3));
    float e0 = expf(v0 - m), e1 = expf(v1 - m), e2 = expf(v2 - m), e3 = expf(v3 - m);
    float inv = 1.0f / (e0 + e1 + e2 + e3);
    float* o = aw + (((long)h * NTOK + n) * SWIN + s) * 4;
    o[0] = e0 * inv; o[1] = e1 * inv; o[2] = e2 * inv; o[3] = e3 * inv;
}

// ---------------- x_diag[n,d] = sum_f attnw[h(d),n,s0,f] * V[f*S+s0, d] ----------------
__global__ void xdiag_kernel(const float* __restrict__ aw, const float* __restrict__ qkv, float* __restrict__ xd) {
    int i = blockIdx.x * 256 + threadIdx.x;
    if (i >= NTOK * DMODEL) return;
    int n = i / DMODEL, d = i % DMODEL;
    int s0 = n & 255; int h = d >> 6;
    const float* a = aw + (((long)h * NTOK + n) * SWIN + s0) * 4;
    float acc = 0.f;
    #pragma unroll
    for (int f = 0; f < FNUM; f++)
        acc += a[f] * qkv[(long)(f * SWIN + s0) * QKVLD + 768 + d];
    xd[i] = acc;
}

// ---------------- fused stage-2 spatial attention (low-rank via P) ----------------
// logits[h,s] = sum_{h',f} attnw[h',n,s,f] * (q2[h] . P[h', f*S+s, h*64:h*64+64])
// o[n,c]     = sum_s w[h(c),s] * sum_{h',f} attnw[h',n,s,f] * P[h', f*S+s, 384+c]
__global__ __launch_bounds__(DMODEL) void stage2_kernel(
    const float* __restrict__ aw_g, const float* __restrict__ q2s,
    const float* __restrict__ P, float* __restrict__ o, float* __restrict__ sattn)
{
    __shared__ float aw[HNUM * SWIN * 4];                 // 24 KB
    __shared__ __align__(16) float q2l[DMODEL];
    __shared__ float qpk[HNUM * NTOK];                    // 24 KB, reused per h
    __shared__ float wgt[HNUM * SWIN];                    // 6 KB
    __shared__ float red[SWIN];
    int n = blockIdx.x, t = threadIdx.x;

    for (int i = t; i < HNUM * SWIN * 4; i += DMODEL) {
        int hp = i >> 10; int r = i & 1023;
        aw[i] = aw_g[(long)hp * (NTOK * SWIN * 4) + (long)n * (SWIN * 4) + r];
    }
    q2l[t] = q2s[(long)n * DMODEL + t];
    __syncthreads();

    for (int h = 0; h < HNUM; h++) {
        // QPk[h'][m] = q2[h] . P[h', m, h*64 : h*64+64]
        for (int e = t; e < HNUM * NTOK; e += DMODEL) {
            int hp = e >> 10; int m = e & 1023;
            const float4* p4 = (const float4*)(P + ((long)((hp << 10) + m)) * 768 + h * 64);
            const float4* q4 = (const float4*)(q2l + h * 64);
            float acc = 0.f;
            #pragma unroll
            for (int j = 0; j < 16; j++) {
                float4 pv = p4[j]; float4 qv = q4[j];
                acc += pv.x * qv.x + pv.y * qv.y + pv.z * qv.z + pv.w * qv.w;
            }
            qpk[e] = acc;
        }
        __syncthreads();
        if (t < SWIN) {
            float l = 0.f;
            #pragma unroll
            for (int hp = 0; hp < HNUM; hp++)
                #pragma unroll
                for (int f = 0; f < FNUM; f++)
                    l += aw[((hp << 8) + t) * 4 + f] * qpk[(hp << 10) + (f << 8) + t];
            wgt[(h << 8) + t] = l; red[t] = l;
        }
        __syncthreads();
        for (int st = 128; st > 0; st >>= 1) { if (t < st) red[t] = fmaxf(red[t], red[t + st]); __syncthreads(); }
        float mx = red[0]; __syncthreads();
        if (t < SWIN) { float e_ = expf(wgt[(h << 8) + t] - mx); wgt[(h << 8) + t] = e_; red[t] = e_; }
        __syncthreads();
        for (int st = 128; st > 0; st >>= 1) { if (t < st) red[t] += red[t + st]; __syncthreads(); }
        float inv = 1.0f / red[0]; __syncthreads();
        if (t < SWIN) {
            float w_ = wgt[(h << 8) + t] * inv;
            wgt[(h << 8) + t] = w_;
            sattn[((long)h * NTOK + n) * SWIN + t] = w_;      // space_attn output
        }
        __syncthreads();
    }

    // pass 2: o accumulation, thread t owns channel c = t (head = t>>6)
    int hme = t >> 6;
    float oacc = 0.f;
    for (int s = 0; s < SWIN; s++) {
        float v2 = 0.f;
        #pragma unroll
        for (int hp = 0; hp < HNUM; hp++) {
            const float* a = &aw[((hp << 8) + s) * 4];
            const float* pb = P + ((long)((hp << 10) + s)) * 768 + 384 + t;
            #pragma unroll
            for (int f = 0; f < FNUM; f++)
                v2 += a[f] * pb[(long)f * (256 * 768)];
        }
        oacc += wgt[(hme << 8) + s] * v2;
    }
    o[(long)n * DMODEL + t] = oacc;
}

// ---------------- bilinear 2x upsample, h2 (N,D) -> out (BF,D,32,32) ----------------
__global__ void upsample_kernel(const float* __restrict__ h2, float* __restrict__ out) {
    int i = blockIdx.x * 256 + threadIdx.x;
    if (i >= FNUM * DMODEL * 32 * 32) return;
    int f = i / (DMODEL * 1024);
    int r = i % (DMODEL * 1024);
    int d = r / 1024;
    int yx = r & 1023; int y = yx >> 5, x = yx & 31;
    int jy = y >> 1, jx = x >> 1;
    int iy0, iy1, ix0, ix1; float wy0, wy1, wx0, wx1;
    if ((y & 1) == 0) { iy0 = max(jy - 1, 0); iy1 = jy;              wy0 = 0.25f; wy1 = 0.75f; }
    else              { iy0 = jy;             iy1 = min(jy + 1, 15); wy0 = 0.75f; wy1 = 0.25f; }
    if ((x & 1) == 0) { ix0 = max(jx - 1, 0); ix1 = jx;              wx0 = 0.25f; wx1 = 0.75f; }
    else              { ix0 = jx;             ix1 = min(jx + 1, 15); wx0 = 0.75f; wx1 = 0.25f; }
    const float* base = h2 + (long)f * SWIN * DMODEL;
    float v = wy0 * (wx0 * base[(iy0 * 16 + ix0) * DMODEL + d] + wx1 * base[(iy0 * 16 + ix1) * DMODEL + d])
            + wy1 * (wx0 * base[(iy1 * 16 + ix0) * DMODEL + d] + wx1 * base[(iy1 * 16 + ix1) * DMODEL + d]);
    out[i] = v;
}

// ---------------- host ----------------
static inline void launch_gemm(hipStream_t stream,
    const float* A, int lda, long sA, const float* B, int ldb, long sB, int transB,
    float* C, int ldc, long sC, const float* bias, const float* resid,
    int M, int N, int K, float alpha, int act, int Z)
{
    int tiles = (M >> 5) * (N >> 5);
    dim3 grid((tiles + 7) / 8, 1, Z), block(256);
    if (transB)
        hipLaunchKernelGGL((gemm_wmma_f32<1>), grid, block, 0, stream,
                           A, lda, sA, B, ldb, sB, C, ldc, sC, bias, resid, M, N, K, alpha, act);
    else
        hipLaunchKernelGGL((gemm_wmma_f32<0>), grid, block, 0, stream,
                           A, lda, sA, B, ldb, sB, C, ldc, sC, bias, resid, M, N, K, alpha, act);
}

extern "C" void kernel_launch(void* const* d_in, const int* in_sizes, int n_in,
                              void* d_out, int out_size, void* d_ws, size_t ws_size,
                              hipStream_t stream) {
    (void)in_sizes; (void)n_in; (void)out_size; (void)ws_size;
    const float* x      = (const float*)d_in[0];
    const float* qkv_w  = (const float*)d_in[1];
    const float* q2_w   = (const float*)d_in[2];
    const float* kv2_w  = (const float*)d_in[3];
    const float* proj_w = (const float*)d_in[4];
    const float* proj_b = (const float*)d_in[5];
    const float* ln1_g  = (const float*)d_in[6];
    const float* ln1_b  = (const float*)d_in[7];
    const float* ln2_g  = (const float*)d_in[8];
    const float* ln2_b  = (const float*)d_in[9];
    const float* fc1_w  = (const float*)d_in[10];
    const float* fc1_b  = (const float*)d_in[11];
    const float* fc2_w  = (const float*)d_in[12];
    const float* fc2_b  = (const float*)d_in[13];
    float* outp  = (float*)d_out;                       // (BF,D,32,32)
    float* sattn = outp + (long)FNUM * DMODEL * 1024;   // (H,N,S)

    float* ws = (float*)d_ws;
    float* tok   = ws;                 // 393216
    float* xn    = tok   + 393216;     // 393216
    float* qkv   = xn    + 393216;     // 1179648
    float* qk    = qkv   + 1179648;    // 6291456
    float* attnw = qk    + 6291456;    // 6291456
    float* Pbuf  = attnw + 6291456;    // 4718592
    float* xdg   = Pbuf  + 4718592;    // 393216
    float* q2s   = xdg   + 393216;     // 393216
    float* obuf  = q2s   + 393216;     // 393216
    float* h1    = obuf  + 393216;     // 393216
    float* hn    = h1    + 393216;     // 393216
    float* mlph  = hn    + 393216;     // 1572864
    float* h2    = mlph  + 1572864;    // 393216  (total ~92.8 MB)

    // 1) maxpool -> tok ; LN1 -> xn
    hipLaunchKernelGGL(patch_kernel, dim3(1536), dim3(256), 0, stream, x, tok);
    hipLaunchKernelGGL(ln_kernel, dim3(NTOK), dim3(DMODEL), 0, stream, tok, xn, ln1_g, ln1_b);

    // 2) qkv = xn @ qkv_w  (1024x384 @ 384x1152)
    launch_gemm(stream, xn, DMODEL, 0, qkv_w, 3 * DMODEL, 0, 0,
                qkv, 3 * DMODEL, 0, nullptr, nullptr, NTOK, 3 * DMODEL, DMODEL, 1.0f, 0, 1);

    // 3) qk[h] = Q_h @ K_h^T   (batched over heads, K=64)
    launch_gemm(stream, qkv, QKVLD, 64, qkv + 384, QKVLD, 64, 1,
                qk, NTOK, (long)NTOK * NTOK, nullptr, nullptr, NTOK, NTOK, HDIM, 1.0f, 0, HNUM);

    // 4) frame softmax -> attnw[h,n,s,f]
    hipLaunchKernelGGL(tsoftmax_kernel, dim3(6144), dim3(256), 0, stream, qk, attnw);

    // 5) P[h] = V_h @ kv2_w_h   (1024x64 @ 64x768, batched)
    launch_gemm(stream, qkv + 768, QKVLD, 64, kv2_w, 2 * DMODEL, (long)64 * 768, 0,
                Pbuf, 2 * DMODEL, (long)NTOK * 768, nullptr, nullptr, NTOK, 2 * DMODEL, HDIM, 1.0f, 0, HNUM);

    // 6) x_diag ; q2 = SCALE * (x_diag @ q2_w)
    hipLaunchKernelGGL(xdiag_kernel, dim3(1536), dim3(256), 0, stream, attnw, qkv, xdg);
    launch_gemm(stream, xdg, DMODEL, 0, q2_w, DMODEL, 0, 0,
                q2s, DMODEL, 0, nullptr, nullptr, NTOK, DMODEL, DMODEL, SCALEF, 0, 1);

    // 7) fused stage-2 attention -> o, space_attn
    hipLaunchKernelGGL(stage2_kernel, dim3(NTOK), dim3(DMODEL), 0, stream, attnw, q2s, Pbuf, obuf, sattn);

    // 8) h1 = tok + o @ proj_w + proj_b
    launch_gemm(stream, obuf, DMODEL, 0, proj_w, DMODEL, 0, 0,
                h1, DMODEL, 0, proj_b, tok, NTOK, DMODEL, DMODEL, 1.0f, 0, 1);

    // 9) LN2 -> hn ; MLP ; h2 = h1 + mlp
    hipLaunchKernelGGL(ln_kernel, dim3(NTOK), dim3(DMODEL), 0, stream, h1, hn, ln2_g, ln2_b);
    launch_gemm(stream, hn, DMODEL, 0, fc1_w, 4 * DMODEL, 0, 0,
                mlph, 4 * DMODEL, 0, fc1_b, nullptr, NTOK, 4 * DMODEL, DMODEL, 1.0f, 1, 1);
    launch_gemm(stream, mlph, 4 * DMODEL, 0, fc2_w, DMODEL, 0, 0,
                h2, DMODEL, 0, fc2_b, h1, NTOK, DMODEL, 4 * DMODEL, 1.0f, 0, 1);

    // 10) bilinear upsample 16->32 -> out
    hipLaunchKernelGGL(upsample_kernel, dim3(6144), dim3(256), 0, stream, h2, outp);
}